// HashEncoding_5136780886372
// MI455X (gfx1250) — compile-verified
//
#include <hip/hip_runtime.h>

typedef float        f32x2 __attribute__((ext_vector_type(2)));
typedef unsigned int u32x4 __attribute__((ext_vector_type(4)));
typedef int          i32x4 __attribute__((ext_vector_type(4)));
typedef int          i32x8 __attribute__((ext_vector_type(8)));

#define HASH_BITS   19
#define TBL_MASK    ((1u << HASH_BITS) - 1u)
#define PRIME1      2654435761u
#define PRIME2      805459861u
#define NLVL        16
#define TPB         256
#define PTS_PER_BLK 256
#define QTAB_TOTAL  2415   // sum of per-level resolutions

#if defined(__has_builtin)
#if __has_builtin(__builtin_amdgcn_tensor_load_to_lds) && __has_builtin(__builtin_amdgcn_s_wait_tensorcnt)
#define USE_TDM 1
#endif
#endif

__global__ __launch_bounds__(TPB) void hashgrid_encode(
    const float* __restrict__ xin, const float* __restrict__ tables,
    float* __restrict__ out, int nPoints)
{
    // int(16 * 32**(i/15)) with IEEE double semantics (note level 9 -> 127, not 128)
    static constexpr int RES[NLVL] = {16,20,25,32,40,50,64,80,101,127,161,203,256,322,406,512};
    static constexpr int QB [NLVL] = {0,16,36,61,93,133,183,247,327,428,555,716,919,1175,1497,1903};

    __shared__ float    sx[PTS_PER_BLK * 3];   // 3 KB coordinate tile (TDM target)
    __shared__ unsigned qtab[QTAB_TOTAL];      // per-(c,res) hash quantization table
    __shared__ int      sres[NLVL];
    __shared__ int      sqb [NLVL];

    const int tid = threadIdx.x;
    const long long blkBase = (long long)blockIdx.x * PTS_PER_BLK;

#ifdef USE_TDM
    // ---- Tensor Data Mover: async bulk copy of this block's x tile into LDS ----
    if (tid == 0) {
        unsigned long long ga = (unsigned long long)(const void*)(xin + blkBase * 3);
        unsigned ldsOff = (unsigned)(unsigned long long)&sx[0];
        unsigned totalF = (unsigned)nPoints * 3u;
        unsigned startF = (unsigned)(blkBase * 3);
        unsigned td0    = totalF - startF; if (td0 > 768u) td0 = 768u; // OOB reads -> 0

        u32x4 g0;
        g0.x = 1u;                                              // count=1 valid, no gather
        g0.y = ldsOff;                                          // lds_addr (bytes)
        g0.z = (unsigned)(ga & 0xFFFFFFFFull);                  // global_addr[31:0]
        g0.w = (unsigned)((ga >> 32) & 0x01FFFFFFull) | (2u << 30); // addr[56:32] | type=2

        i32x8 g1;
        g1[0] = 0x00020000;                          // workgroup_mask=0, data_size=2 (4B)
        g1[1] = (int)((td0 & 0xFFFFu) << 16);        // tensor_dim0 low16 @ bits63:48
        g1[2] = (int)(((td0 >> 16) & 0xFFFFu) | (1u << 16)); // dim0 hi16 | tensor_dim1=1
        g1[3] = (int)(768u << 16);                   // tile_dim0 = 768 elements
        g1[4] = 1;                                   // tile_dim1 = 1, tile_dim2 = 0
        g1[5] = 768;                                 // tensor_dim0_stride low32
        g1[6] = 0;
        g1[7] = 0;
        i32x4 z4 = {0, 0, 0, 0};
        i32x8 z8 = {0, 0, 0, 0, 0, 0, 0, 0};
        // clang-23 / therock-10.0 arity: (g0 v4u, g1 v8i, g2 v4i, g3 v4i, v8i, i32 cpol)
        __builtin_amdgcn_tensor_load_to_lds(g0, g1, z4, z4, z8, 0);
    }
#else
    {
        const int startF = (int)(blkBase * 3);
        const int totalF = nPoints * 3;
        for (int i = tid; i < PTS_PER_BLK * 3; i += TPB) {
            int g = startF + i;
            sx[i] = (g < totalF) ? xin[g] : 0.f;
        }
    }
#endif

    // ---- Build q-table: q = trunc((c/res + 1.0f) * 2^17), bit-exact IEEE f32 path ----
#pragma unroll
    for (int l = 0; l < NLVL; ++l) {
        const int r = RES[l];
        const int b = QB[l];
        for (int c = tid; c < r; c += TPB) {
            float cf = (float)c / (float)r;            // IEEE RN division (matches ref)
            qtab[b + c] = (unsigned)((cf + 1.0f) * 131072.0f);
        }
    }
    if (tid == 0) {
#pragma unroll
        for (int l = 0; l < NLVL; ++l) { sres[l] = RES[l]; sqb[l] = QB[l]; }
#ifdef USE_TDM
        __builtin_amdgcn_s_wait_tensorcnt(0);          // x tile resident in LDS
#endif
    }
    __syncthreads();

    // ---- Main loop: lane = (level); 16 lanes per point; 16 points per thread ----
    const int level = tid & 15;
    const int grp   = tid >> 4;
    const int res   = sres[level];
    const int qb    = sqb[level];
    const float fres = (float)res;
    const f32x2* __restrict__ tbl = (const f32x2*)tables + ((size_t)level << HASH_BITS);

#pragma unroll 2
    for (int it = 0; it < 16; ++it) {
        const int pl = it * 16 + grp;
        const long long pg = blkBase + pl;
        if (pg >= (long long)nPoints) continue;

        float vx = sx[pl * 3 + 0];
        float vy = sx[pl * 3 + 1];
        float vz = sx[pl * 3 + 2];
        vx = fminf(fmaxf(vx, -1.f), 1.f);
        vy = fminf(fmaxf(vy, -1.f), 1.f);
        vz = fminf(fmaxf(vz, -1.f), 1.f);

        // dim 0 (prime = 1)
        float crd0 = vx * fres, flr0 = floorf(crd0), lc0 = crd0 - flr0;
        int i00 = (int)flr0; if (i00 < 0) i00 += res; if (i00 >= res) i00 -= res;
        int i01 = i00 + 1;   if (i01 >= res) i01 = 0;
        unsigned hx0 = qtab[qb + i00];
        unsigned hx1 = qtab[qb + i01];
        // dim 1 (prime = PRIME1)
        float crd1 = vy * fres, flr1 = floorf(crd1), lc1 = crd1 - flr1;
        int i10 = (int)flr1; if (i10 < 0) i10 += res; if (i10 >= res) i10 -= res;
        int i11 = i10 + 1;   if (i11 >= res) i11 = 0;
        unsigned hy0 = qtab[qb + i10] * PRIME1;
        unsigned hy1 = qtab[qb + i11] * PRIME1;
        // dim 2 (prime = PRIME2)
        float crd2 = vz * fres, flr2 = floorf(crd2), lc2 = crd2 - flr2;
        int i20 = (int)flr2; if (i20 < 0) i20 += res; if (i20 >= res) i20 -= res;
        int i21 = i20 + 1;   if (i21 >= res) i21 = 0;
        unsigned hz0 = qtab[qb + i20] * PRIME2;
        unsigned hz1 = qtab[qb + i21] * PRIME2;

        // 8 corner hashes, corner k = (bx<<2)|(by<<1)|bz  (matches meshgrid 'ij' offsets)
        unsigned s00 = hx0 + hy0, s01 = hx0 + hy1, s10 = hx1 + hy0, s11 = hx1 + hy1;
        unsigned h0 = (s00 + hz0) & TBL_MASK, h1 = (s00 + hz1) & TBL_MASK;
        unsigned h2 = (s01 + hz0) & TBL_MASK, h3 = (s01 + hz1) & TBL_MASK;
        unsigned h4 = (s10 + hz0) & TBL_MASK, h5 = (s10 + hz1) & TBL_MASK;
        unsigned h6 = (s11 + hz0) & TBL_MASK, h7 = (s11 + hz1) & TBL_MASK;

        // 8 x global_load_b64 gathers (tables stay L2-resident; ~64MB < 192MB L2)
        f32x2 f0 = tbl[h0], f1 = tbl[h1], f2 = tbl[h2], f3 = tbl[h3];
        f32x2 f4 = tbl[h4], f5 = tbl[h5], f6 = tbl[h6], f7 = tbl[h7];

        // trilinear weights, product order ((wx*wy)*wz) like reference prod(-1)
        float wx0 = 1.f - lc0, wx1 = lc0;
        float wy0 = 1.f - lc1, wy1 = lc1;
        float wz0 = 1.f - lc2, wz1 = lc2;
        float p00 = wx0 * wy0, p01 = wx0 * wy1, p10 = wx1 * wy0, p11 = wx1 * wy1;
        float w0 = p00 * wz0, w1 = p00 * wz1, w2 = p01 * wz0, w3 = p01 * wz1;
        float w4 = p10 * wz0, w5 = p10 * wz1, w6 = p11 * wz0, w7 = p11 * wz1;

        float a0 = w0 * f0.x; float a1 = w0 * f0.y;
        a0 += w1 * f1.x; a1 += w1 * f1.y;
        a0 += w2 * f2.x; a1 += w2 * f2.y;
        a0 += w3 * f3.x; a1 += w3 * f3.y;
        a0 += w4 * f4.x; a1 += w4 * f4.y;
        a0 += w5 * f5.x; a1 += w5 * f5.y;
        a0 += w6 * f6.x; a1 += w6 * f6.y;
        a0 += w7 * f7.x; a1 += w7 * f7.y;

        f32x2 ov; ov.x = a0; ov.y = a1;
        // NT store: 128MB output stream must not evict the L2-resident tables
        __builtin_nontemporal_store(ov, (f32x2*)(out + (size_t)pg * 32 + (level << 1)));
    }
}

extern "C" void kernel_launch(void* const* d_in, const int* in_sizes, int n_in,
                              void* d_out, int out_size, void* d_ws, size_t ws_size,
                              hipStream_t stream) {
    (void)n_in; (void)out_size; (void)d_ws; (void)ws_size;
    const float* x      = (const float*)d_in[0];
    const float* tables = (const float*)d_in[1];
    float*       out    = (float*)d_out;
    const int nPoints = in_sizes[0] / 3;
    const int blocks  = (nPoints + PTS_PER_BLK - 1) / PTS_PER_BLK;
    hipLaunchKernelGGL(hashgrid_encode, dim3(blocks), dim3(TPB), 0, stream,
                       x, tables, out, nPoints);
}